// ExponentialMovingAverage_56891136803334
// MI455X (gfx1250) — compile-verified
//
#include <hip/hip_runtime.h>
#include <hip/hip_bf16.h>
#include <stdint.h>

typedef float v2f __attribute__((ext_vector_type(2)));
typedef float v8f __attribute__((ext_vector_type(8)));

// Problem shape (from reference setup_inputs)
constexpr int B = 16, T = 8192, F = 512;
constexpr int S = 16;               // time segments
constexpr int TSEG = T / S;         // 512 timesteps per segment
constexpr int CH = 16;              // timesteps per WMMA chunk
constexpr int NCHUNK = TSEG / CH;   // 32 chunks per segment
constexpr int FBLK = 128;           // f-columns per block (8 waves x 16)
constexpr int RS = FBLK + 16;       // padded LDS row stride (floats) -> bank-conflict-free

// ---------------------------------------------------------------------------
// K1: per-(b,seg,f) weighted reduction  s = sum_t w * d^(L-1-t) * x_t
//     (forward form: s = d*s + w*x_t). Reads x once, coalesced 1KB/row.
// ---------------------------------------------------------------------------
__global__ __launch_bounds__(256) void ema_k1_segsum(
    const float* __restrict__ x, const float* __restrict__ smooth,
    float* __restrict__ segsum /* [B][S][F] */) {
  const int blk = blockIdx.x;          // B * S * 2 blocks
  const int fgrp = blk & 1;
  const int seg = (blk >> 1) & (S - 1);
  const int b = blk >> 5;
  const int f = fgrp * 256 + threadIdx.x;

  const float w = fminf(fmaxf(smooth[0], 0.f), 1.f);
  const float d = 1.f - w;

  const float* xp = x + ((size_t)b * T + (size_t)seg * TSEG) * F + f;
  float s = 0.f;
#pragma unroll 8
  for (int t = 0; t < TSEG; ++t) s = fmaf(d, s, w * xp[(size_t)t * F]);
  segsum[((size_t)b * S + seg) * F + f] = s;
}

// ---------------------------------------------------------------------------
// K2: tiny sequential scan over S=16 segment contributions per (b,f).
//     carry[b][k][f] = state ENTERING segment k.
// ---------------------------------------------------------------------------
__global__ __launch_bounds__(256) void ema_k2_scan(
    const float* __restrict__ init, const float* __restrict__ smooth,
    const float* __restrict__ segsum, float* __restrict__ carry) {
  const int idx = blockIdx.x * 256 + threadIdx.x;  // b*F + f
  const int b = idx >> 9;
  const int f = idx & (F - 1);

  const float w = fminf(fmaxf(smooth[0], 0.f), 1.f);
  float dL = 1.f - w;
  for (int i = 0; i < 9; ++i) dL *= dL;  // d^512 by squaring (TSEG = 2^9)

  float a = init[idx];
#pragma unroll
  for (int k = 0; k < S; ++k) {
    const size_t o = ((size_t)b * S + k) * F + f;
    carry[o] = a;
    a = fmaf(dL, a, segsum[o]);
  }
}

// ---------------------------------------------------------------------------
// K3: scan-as-matmul. Per wave: 16 f-columns x one segment. For each 16-step
//     chunk:  Y(16x16) = L(16x16) * X(16x16) + dpow(+)carry  via 4 chained
//     v_wmma_f32_16x16x4_f32. X staged into LDS with async b128 loads.
// ---------------------------------------------------------------------------
__global__ __launch_bounds__(256) void ema_k3_wmma(
    const float* __restrict__ x, const float* __restrict__ smooth,
    const float* __restrict__ carry_ws, float* __restrict__ y) {
  __shared__ float lds[2][CH * RS];  // double-buffered 16x128 slab (padded)

  const int tid = threadIdx.x;
  const int lane = tid & 31;
  const int wv = tid >> 5;        // wave 0..7 -> f-subtile
  const int m = lane & 15;        // row (M) / column (N) index within tile
  const int hi = lane >> 4;       // half-wave select

  const int fblk = blockIdx.x & 3;
  const int seg = (blockIdx.x >> 2) & (S - 1);
  const int b = blockIdx.x >> 6;

  const float w = fminf(fmaxf(smooth[0], 0.f), 1.f);
  const float d = 1.f - w;

  auto dpow = [&](int e) {
    float p = 1.f;
    for (int i = 0; i < e; ++i) p *= d;
    return p;
  };

  // A operand = L (lower-triangular decay matrix), constant per lane.
  // 32-bit A 16x4 layout: VGPR0 = {K=0 | K=2}, VGPR1 = {K=1 | K=3}.
  v2f A[4];
#pragma unroll
  for (int k = 0; k < 4; ++k) {
    const int j0 = 4 * k + (hi ? 2 : 0);
    const int j1 = 4 * k + (hi ? 3 : 1);
    A[k].x = (j0 <= m) ? w * dpow(m - j0) : 0.f;
    A[k].y = (j1 <= m) ? w * dpow(m - j1) : 0.f;
  }
  // C scale factors: C[row][n] = d^(row+1) * carry[n]; VGPR v -> row v + hi*8.
  float dpw[8];
  {
    float p = d * dpow(hi * 8);
#pragma unroll
    for (int v = 0; v < 8; ++v) { dpw[v] = p; p *= d; }
  }

  const float* xseg = x + ((size_t)b * T + (size_t)seg * TSEG) * F + fblk * FBLK;
  float* yseg = y + ((size_t)b * T + (size_t)seg * TSEG) * F + fblk * FBLK;

  // state entering this segment for this lane's column (replicated half-waves)
  float carry = carry_ws[((size_t)b * S + seg) * F + fblk * FBLK + wv * 16 + m];

  // Async slab loader: 16 rows x 512B -> 512 b128 units; 2 units per thread.
  auto issue = [&](int chunk, int buf) {
    const float* gbase = xseg + (size_t)chunk * CH * F;
#pragma unroll
    for (int q = 0; q < 2; ++q) {
      const int u = tid + q * 256;
      const int r = u >> 5;            // row 0..15
      const int c4 = (u & 31) << 2;    // col 0..124 step 4
      const float* gp = gbase + (size_t)r * F + c4;
      // LDS byte offset = low 32 bits of flat shared address (ISA 10.2)
      const unsigned lo = (unsigned)(uintptr_t)(&lds[buf][r * RS + c4]);
      asm volatile("global_load_async_to_lds_b128 %0, %1, off"
                   :: "v"(lo), "v"(gp)
                   : "memory");
    }
  };

  issue(0, 0);

  for (int chunk = 0; chunk < NCHUNK; ++chunk) {
    const int buf = chunk & 1;
    if (chunk + 1 < NCHUNK) {
      issue(chunk + 1, buf ^ 1);
      asm volatile("s_wait_asynccnt 2" ::: "memory");  // current slab landed
    } else {
      asm volatile("s_wait_asynccnt 0" ::: "memory");
    }
    __syncthreads();  // all 8 waves' portions of this slab visible

    const float* sl = &lds[buf][0];
    const int fw = wv * 16 + m;  // this lane's column within the slab

    v8f D;
#pragma unroll
    for (int v = 0; v < 8; ++v) D[v] = dpw[v] * carry;

#pragma unroll
    for (int k = 0; k < 4; ++k) {
      // B 4x16 layout (mirrors A): VGPR0 = {K=0 | K=2}, VGPR1 = {K=1 | K=3}
      const int r0 = 4 * k + (hi ? 2 : 0);
      const int r1 = 4 * k + (hi ? 3 : 1);
      v2f Bv;
      Bv.x = sl[r0 * RS + fw];
      Bv.y = sl[r1 * RS + fw];
      D = __builtin_amdgcn_wmma_f32_16x16x4_f32(
          /*neg_a=*/false, A[k], /*neg_b=*/false, Bv,
          /*c_mod=*/(short)0, D, /*reuse_a=*/false, /*reuse_b=*/false);
    }

    // Store D: VGPR v holds row (t0 + v + hi*8), col f0 + m.
    float* yo = yseg + (size_t)chunk * CH * F + wv * 16 + m;
#pragma unroll
    for (int v = 0; v < 8; ++v) yo[(size_t)(v + hi * 8) * F] = D[v];

    // New carry = row 15 of D = VGPR7 on lanes 16..31; replicate to both halves.
    carry = __shfl(D[7], 16 + m, 32);

    __syncthreads();  // everyone done reading buf before it is refilled
  }
}

extern "C" void kernel_launch(void* const* d_in, const int* in_sizes, int n_in,
                              void* d_out, int out_size, void* d_ws, size_t ws_size,
                              hipStream_t stream) {
  (void)in_sizes; (void)n_in; (void)out_size; (void)ws_size;
  const float* x = (const float*)d_in[0];
  const float* init = (const float*)d_in[1];
  const float* smooth = (const float*)d_in[2];
  float* y = (float*)d_out;

  float* segsum = (float*)d_ws;                       // B*S*F floats (512 KB)
  float* carry = segsum + (size_t)B * S * F;          // B*S*F floats (512 KB)

  ema_k1_segsum<<<B * S * (F / 256), 256, 0, stream>>>(x, smooth, segsum);
  ema_k2_scan<<<(B * F) / 256, 256, 0, stream>>>(init, smooth, segsum, carry);
  ema_k3_wmma<<<B * S * (F / FBLK), 256, 0, stream>>>(x, smooth, carry, y);
}